// EnhancedGNNEncoder_43593918054657
// MI455X (gfx1250) — compile-verified
//
#include <hip/hip_runtime.h>
#include <hip/hip_bf16.h>
#include <math.h>

// ---------------------------------------------------------------------------
// EnhancedGNNEncoder for MI455X (gfx1250, wave32).
//  - Edge phase: 1 wave per edge, ds_swizzle-reduced 32-dot-products,
//    hardware global_atomic_add_f32 scatter into an L2-resident accumulator.
//  - Node phase + final FC: fp32 WMMA (v_wmma_f32_16x16x4_f32) 128x128 GEMMs
//    with fused ReLU + LayerNorm + residual.
// ---------------------------------------------------------------------------

typedef __attribute__((ext_vector_type(2))) float v2f;
typedef __attribute__((ext_vector_type(8))) float v8f;

#define HID    128
#define EDIM   32
#define NTYPES 2
#define ETYPES 3
#define NLAYER 3
#define PUMP_T 1
#define LN_EPS 1e-5f

// ds_swizzle group-of-32 XOR shuffle (single instruction on wave32)
template <int M>
__device__ __forceinline__ float xorswz(float x) {
  return __int_as_float(
      __builtin_amdgcn_ds_swizzle(__float_as_int(x), (M << 10) | 0x1f));
}
__device__ __forceinline__ float bcast(float x, int lane) {
  return __int_as_float(__builtin_amdgcn_readlane(__float_as_int(x), lane));
}

// ---------------------------------------------------------------- utilities
__global__ void zero_kernel(float* __restrict__ p, int n4) {
  int i = blockIdx.x * blockDim.x + threadIdx.x;
  if (i < n4) ((float4*)p)[i] = make_float4(0.f, 0.f, 0.f, 0.f);
}

// Pack W[m][o][k] (128x128 row-major) into K-pair-major form:
//   Wt[m][k>>1][o][k&1]  -> a lane's B-fragment {B[k][o], B[k+1][o]} is one
//   contiguous 8B load, coalesced across lanes (consecutive o).
__global__ void pack_weights_kernel(const float* __restrict__ W,
                                    float* __restrict__ Wt, int nmats) {
  int i = blockIdx.x * blockDim.x + threadIdx.x;
  if (i >= nmats * 128 * 128) return;
  int m = i >> 14;
  int o = (i >> 7) & 127;
  int k = i & 127;
  Wt[(m << 14) + (k >> 1) * 256 + o * 2 + (k & 1)] = W[i];
}

// ---------------------------------------------------------------- edge phase
// One wave32 per edge. lane = edge-feature channel for the scalar part,
// then lane handles channels [4*lane, 4*lane+4) of the 128-dim message.
__global__ void __launch_bounds__(256) edge_kernel(
    const float* __restrict__ h,
    const float* __restrict__ edge_attr,   // [E][32]
    const int*   __restrict__ edge_index,  // [2][E] flat
    const int*   __restrict__ edge_type,   // [E]
    const float* __restrict__ eW_l,        // [ET][2][32]
    const float* __restrict__ eb_l,        // [ET][2]
    const float* __restrict__ emb_l,       // [ET][32]
    float* __restrict__ aggr,              // [N][128]
    int E)
{
  int tid  = blockIdx.x * blockDim.x + threadIdx.x;
  int e    = tid >> 5;
  int lane = tid & 31;
  if (e >= E) return;  // uniform per wave

  int   et   = edge_type[e];
  float raw  = edge_attr[e * EDIM + lane];  // coalesced 128B row
  float dir  = bcast(raw, 30);              // edge_attr[:, -2]
  float pump = bcast(raw, 31);              // edge_attr[:, -1]

  float a  = raw + emb_l[et * EDIM + lane];  // ea = edge_attr + emb[et]
  float r0 = a * eW_l[(et * 2 + 0) * EDIM + lane];
  float r1 = a * eW_l[(et * 2 + 1) * EDIM + lane];
  r0 += xorswz<16>(r0); r1 += xorswz<16>(r1);
  r0 += xorswz<8>(r0);  r1 += xorswz<8>(r1);
  r0 += xorswz<4>(r0);  r1 += xorswz<4>(r1);
  r0 += xorswz<2>(r0);  r1 += xorswz<2>(r1);
  r0 += xorswz<1>(r0);  r1 += xorswz<1>(r1);
  r0 += eb_l[et * 2 + 0];
  r1 += eb_l[et * 2 + 1];

  // softplus, numerically stable
  float gain  = fmaxf(r0, 0.f) + log1pf(expf(-fabsf(r0)));
  float speed = pump * (dir > 0.f ? dir : 1.f);
  bool  is_pump = (et == PUMP_T);
  if (is_pump) gain *= speed;
  float bias = is_pump ? r1 * speed : 0.f;
  float sgn  = dir * 2.f - 1.f;

  int src = edge_index[e];
  int dst = edge_index[E + e];

  const float4 hs = *(const float4*)(h + (size_t)src * HID + lane * 4);
  const float4 hd = *(const float4*)(h + (size_t)dst * HID + lane * 4);
  float4 msg;
  msg.x = sgn * (gain * (hs.x - hd.x) + bias);
  msg.y = sgn * (gain * (hs.y - hd.y) + bias);
  msg.z = sgn * (gain * (hs.z - hd.z) + bias);
  msg.w = sgn * (gain * (hs.w - hd.w) + bias);

  float* out = aggr + (size_t)dst * HID + lane * 4;
  unsafeAtomicAdd(out + 0, msg.x);  // global_atomic_add_f32 -> 192MB L2
  unsafeAtomicAdd(out + 1, msg.y);
  unsafeAtomicAdd(out + 2, msg.z);
  unsafeAtomicAdd(out + 3, msg.w);
}

// ---------------------------------------------------------------- node phase
// Block = 256 threads = 8 waves. Block owns 16 rows; wave w owns output
// columns [16w, 16w+16). K = 128 -> 32 WMMA f32 16x16x4 steps.
// A frag (16x4 f32): lane holds A[M=lane&15][K=2*(lane>>4)+v], v=0..1
// B frag (4x16):     lane holds B[K=2*(lane>>4)+v][N=lane&15]
// C/D (16x16 f32):   VGPR i holds row M = i + 8*(lane>=16), col N = lane&15
__global__ void __launch_bounds__(256) node_kernel(
    const float* __restrict__ aggr,
    const float* __restrict__ Wt,        // [NT] packed (pack_weights_kernel)
    const float* __restrict__ nb,        // [NT][128]
    const int*   __restrict__ node_type,
    const float* __restrict__ ln_g,      // [128]
    const float* __restrict__ ln_b,      // [128]
    const float* __restrict__ h_in,      // identity (residual)
    float* __restrict__ h_out,
    int N)
{
  __shared__ float As[16][132];          // pad 132: conflict-free b64 reads
  __shared__ float psum[16][8];
  __shared__ float psum2[16][8];
  __shared__ float rowstat[16][2];
  __shared__ int   rtype[16];

  int tid  = threadIdx.x;
  int row0 = blockIdx.x * 16;

  for (int j = tid; j < 16 * HID; j += 256) {
    int r = j >> 7, c = j & 127;
    int row = row0 + r;
    As[r][c] = (row < N) ? aggr[(size_t)row * HID + c] : 0.f;
  }
  if (tid < 16) {
    int row = row0 + tid;
    rtype[tid] = (row < N) ? node_type[row] : 0;
  }
  __syncthreads();

  int wave  = tid >> 5;
  int lane  = tid & 31;
  int nlo   = lane & 15;
  int khalf = (lane >> 4) * 2;
  int col   = wave * 16 + nlo;

  const float* Wt0 = Wt;
  const float* Wt1 = Wt + 128 * 128;

  v8f acc0 = {};
  v8f acc1 = {};
#pragma unroll 4
  for (int kb = 0; kb < 32; ++kb) {
    int k0 = kb * 4 + khalf;            // even
    int bo = (k0 >> 1) * 256 + col * 2; // packed K-pair offset
    v2f a, b0, b1;
    const float2 av  = *(const float2*)&As[nlo][k0];   // ds_load_b64
    const float2 b0v = *(const float2*)(Wt0 + bo);     // global_load_b64
    const float2 b1v = *(const float2*)(Wt1 + bo);
    a[0] = av.x;  a[1] = av.y;
    b0[0] = b0v.x; b0[1] = b0v.y;
    b1[0] = b1v.x; b1[1] = b1v.y;
    acc0 = __builtin_amdgcn_wmma_f32_16x16x4_f32(false, a, false, b0,
                                                 (short)0, acc0, false, false);
    acc1 = __builtin_amdgcn_wmma_f32_16x16x4_f32(false, a, false, b1,
                                                 (short)0, acc1, false, false);
  }

  // select per-row type result, + bias, ReLU, per-row partial LN sums
  float val[8];
#pragma unroll
  for (int i = 0; i < 8; ++i) {
    int rl = i + ((lane >> 4) << 3);
    int t  = rtype[rl];
    float v = ((t == 0) ? acc0[i] : acc1[i]) + nb[t * HID + col];
    v = fmaxf(v, 0.f);
    val[i] = v;
    float sv = v, sq = v * v;
    sv += xorswz<8>(sv); sq += xorswz<8>(sq);   // 16-lane half-wave sum
    sv += xorswz<4>(sv); sq += xorswz<4>(sq);
    sv += xorswz<2>(sv); sq += xorswz<2>(sq);
    sv += xorswz<1>(sv); sq += xorswz<1>(sq);
    if (nlo == 0) { psum[rl][wave] = sv; psum2[rl][wave] = sq; }
  }
  __syncthreads();

  if (tid < 16) {
    float sv = 0.f, sq = 0.f;
#pragma unroll
    for (int w = 0; w < 8; ++w) { sv += psum[tid][w]; sq += psum2[tid][w]; }
    float mu  = sv * (1.f / HID);
    float var = sq * (1.f / HID) - mu * mu;
    rowstat[tid][0] = mu;
    rowstat[tid][1] = rsqrtf(var + LN_EPS);
  }
  __syncthreads();

  float g = ln_g[col], bb = ln_b[col];
#pragma unroll
  for (int i = 0; i < 8; ++i) {
    int rl  = i + ((lane >> 4) << 3);
    int row = row0 + rl;
    if (row < N) {
      float o = (val[i] - rowstat[rl][0]) * rowstat[rl][1] * g + bb
              + h_in[(size_t)row * HID + col];
      h_out[(size_t)row * HID + col] = o;
    }
  }
}

// ---------------------------------------------------------------- final FC
__global__ void __launch_bounds__(256) fc_kernel(
    const float* __restrict__ h,
    const float* __restrict__ Wt,    // packed (pack_weights_kernel)
    const float* __restrict__ bias,  // [128]
    float* __restrict__ out, int N)
{
  __shared__ float As[16][132];
  int tid  = threadIdx.x;
  int row0 = blockIdx.x * 16;

  for (int j = tid; j < 16 * HID; j += 256) {
    int r = j >> 7, c = j & 127;
    int row = row0 + r;
    As[r][c] = (row < N) ? h[(size_t)row * HID + c] : 0.f;
  }
  __syncthreads();

  int wave  = tid >> 5;
  int lane  = tid & 31;
  int nlo   = lane & 15;
  int khalf = (lane >> 4) * 2;
  int col   = wave * 16 + nlo;

  v8f acc = {};
#pragma unroll 4
  for (int kb = 0; kb < 32; ++kb) {
    int k0 = kb * 4 + khalf;
    v2f a, b;
    const float2 av = *(const float2*)&As[nlo][k0];
    const float2 bv = *(const float2*)(Wt + (k0 >> 1) * 256 + col * 2);
    a[0] = av.x; a[1] = av.y;
    b[0] = bv.x; b[1] = bv.y;
    acc = __builtin_amdgcn_wmma_f32_16x16x4_f32(false, a, false, b,
                                                (short)0, acc, false, false);
  }

  float bb = bias[col];
#pragma unroll
  for (int i = 0; i < 8; ++i) {
    int row = row0 + i + ((lane >> 4) << 3);
    if (row < N) out[(size_t)row * HID + col] = acc[i] + bb;
  }
}

// ---------------------------------------------------------------- launcher
extern "C" void kernel_launch(void* const* d_in, const int* in_sizes, int n_in,
                              void* d_out, int out_size, void* d_ws, size_t ws_size,
                              hipStream_t stream) {
  const float* x         = (const float*)d_in[0];
  const float* edge_attr = (const float*)d_in[1];
  const float* node_W    = (const float*)d_in[2];   // [L][NT][128][128]
  const float* node_b    = (const float*)d_in[3];   // [L][NT][128]
  const float* edge_W    = (const float*)d_in[4];   // [L][ET][2][32]
  const float* edge_b    = (const float*)d_in[5];   // [L][ET][2]
  const float* emb       = (const float*)d_in[6];   // [L][ET][32]
  const float* ln_g      = (const float*)d_in[7];   // [L][128]
  const float* ln_b      = (const float*)d_in[8];   // [L][128]
  const float* fc_W      = (const float*)d_in[9];   // [128][128]
  const float* fc_b      = (const float*)d_in[10];  // [128]
  const int*  edge_index = (const int*)d_in[11];    // [2][E]
  const int*  node_type  = (const int*)d_in[12];
  const int*  edge_type  = (const int*)d_in[13];

  const int N = in_sizes[0] / HID;
  const int E = in_sizes[13];

  // workspace layout (floats): aggr | hA | hB | WtN(L*NT mats) | WtF(1 mat)
  float* ws   = (float*)d_ws;
  size_t nh   = (size_t)N * HID;
  float* aggr = ws;
  float* hA   = aggr + nh;
  float* hB   = hA + nh;
  float* WtN  = hB + nh;
  float* WtF  = WtN + (size_t)NLAYER * NTYPES * 128 * 128;

  {
    int tot = NLAYER * NTYPES * 128 * 128;
    pack_weights_kernel<<<(tot + 255) / 256, 256, 0, stream>>>(node_W, WtN,
                                                               NLAYER * NTYPES);
    pack_weights_kernel<<<(128 * 128 + 255) / 256, 256, 0, stream>>>(fc_W, WtF, 1);
  }

  const float* hin = x;
  float* houts[NLAYER] = {hA, hB, hA};
  int n4      = (int)(nh / 4);
  int zblocks = (n4 + 255) / 256;
  int eblocks = (int)(((long long)E * 32 + 255) / 256);
  int nblocks = (N + 15) / 16;

  for (int l = 0; l < NLAYER; ++l) {
    zero_kernel<<<zblocks, 256, 0, stream>>>(aggr, n4);
    edge_kernel<<<eblocks, 256, 0, stream>>>(
        hin, edge_attr, edge_index, edge_type,
        edge_W + (size_t)l * ETYPES * 2 * EDIM,
        edge_b + (size_t)l * ETYPES * 2,
        emb    + (size_t)l * ETYPES * EDIM,
        aggr, E);
    node_kernel<<<nblocks, 256, 0, stream>>>(
        aggr, WtN + (size_t)l * NTYPES * 128 * 128,
        node_b + (size_t)l * NTYPES * HID, node_type,
        ln_g + (size_t)l * HID, ln_b + (size_t)l * HID,
        hin, houts[l], N);
    hin = houts[l];
  }
  fc_kernel<<<nblocks, 256, 0, stream>>>(hin, WtF, fc_b, (float*)d_out, N);
}